// SineGen_6545530159640
// MI455X (gfx1250) — compile-verified
//
#include <hip/hip_runtime.h>

// ---------------------------------------------------------------------------
// SineGen (NSF source module) for MI455X / gfx1250.
//
// Closed form: the reference's wrap/shift terms are exact integers, so
//   sine[b, t*upp+u, d] = sin(2*pi * frac( base[b,t,d] + (u+1)*rad[b,t,d] ))
// with rad[b,t,d] = frac(f0[b,t]*(d+1)/SR) (+ rand_ini at t==0) and
//   base[b,t,d] = frac( sum_{t'<t} upp*rad[b,t',d] )  (fp64 prefix, tiny).
//
// Roofline: read noise 118MB + write sine 118MB + uv 13MB + noise 118MB
// = ~367MB -> ~16us at 23.3 TB/s: pure streaming. Each (b,frame) owns a
// CONTIGUOUS 2304-float chunk of every [B,L,9] tensor -> one workgroup per
// frame, float4 granularity, b128 nontemporal stores. noise_z is staged
// via gfx1250 async global->LDS (b128) overlapped with the sine math; each
// thread reads back only what it loaded, so a single s_wait_asynccnt 0
// (no cross-wave barrier) covers the noise data.
// ---------------------------------------------------------------------------

#define DIMH 9                      // harmonic_num + 1
#define SRF  24000.0f               // sampling rate
#define SINE_AMP 0.1f
#define NOISE_STD 0.003f

typedef float v4f __attribute__((ext_vector_type(4)));

// Pointee type of the async-LDS builtins (per hipcc diagnostic):
// 'int __vector(4)' in AS1 (global) / AS3 (LDS).
typedef int v4i __attribute__((vector_size(4 * sizeof(int))));
typedef __attribute__((address_space(1))) v4i*  g_v4i_p;
typedef __attribute__((address_space(3))) v4i*  l_v4i_p;
typedef __attribute__((address_space(1))) void* g_void_p;
typedef __attribute__((address_space(3))) void* l_void_p;
typedef __attribute__((address_space(3))) float lds_f32;

__device__ __forceinline__ void async_copy_b128(const float* gsrc, float* ldst) {
#if __has_builtin(__builtin_amdgcn_global_load_async_to_lds_b128)
  __builtin_amdgcn_global_load_async_to_lds_b128(
      (g_v4i_p)(g_void_p)gsrc, (l_v4i_p)(l_void_p)ldst, 0, 0);
#else
  asm volatile("global_load_async_to_lds_b128 %0, %1, off"
               :: "v"((lds_f32*)ldst),
                  "v"((g_void_p)gsrc)
               : "memory");
#endif
}

__device__ __forceinline__ void wait_async0() {
#if __has_builtin(__builtin_amdgcn_s_wait_asynccnt)
  __builtin_amdgcn_s_wait_asynccnt(0);
#else
  asm volatile("s_wait_asynccnt 0" ::: "memory");
#endif
}

// --- Kernel A: per-(b,d) frame-base phases (800-step fp64 prefix, 144 rows) -
__global__ void sine_precompute(const float* __restrict__ f0,
                                const float* __restrict__ rand_ini,
                                float* __restrict__ base,   // [B*T][DIM]
                                int B, int T, int upp) {
  int i = blockIdx.x * blockDim.x + threadIdx.x;
  if (i >= B * DIMH) return;
  int b = i / DIMH;
  int d = i - b * DIMH;
  const float hm = (float)(d + 1);
  double acc = 0.0;
  for (int t = 0; t < T; ++t) {
    // identical op sequence to the reference: mul, div, mod-1
    float r = (f0[b * T + t] * hm) / SRF;
    r = r - floorf(r);
    if (t == 0) r += rand_ini[i];
    base[(long)(b * T + t) * DIMH + d] = (float)(acc - floor(acc));
    acc += (double)r * (double)upp;
  }
}

// --- Kernel B: one workgroup per (b, frame); contiguous 2304-float chunk ----
__global__ void sine_main(const float* __restrict__ f0,
                          const float* __restrict__ rand_ini,
                          const float* __restrict__ noise_z,
                          const float* __restrict__ base,
                          float* __restrict__ sine_out,
                          float* __restrict__ uv_out,
                          float* __restrict__ noise_out,
                          int T, int upp) {
  extern __shared__ __align__(16) float lbuf[];   // upp*DIM floats (9216 B)
  __shared__ float sr[DIMH];                      // per-d rad (div done once)
  __shared__ float sb[DIMH];                      // per-d frame base phase

  const int bt    = blockIdx.x;                   // = b*T + t
  const int t     = bt % T;
  const int b     = bt / T;
  const int tid   = threadIdx.x;
  const int nth   = blockDim.x;                   // = chunk/8 (288)
  const int chunk = upp * DIMH;
  const int nv4   = chunk >> 2;                   // float4 count (576)
  const long foff = (long)bt * chunk;             // contiguous chunk offset

  // 1) kick off async b128 staging of this frame's noise into LDS
  const float* gsrc = noise_z + foff;
  for (int v = tid; v < nv4; v += nth)
    async_copy_b128(gsrc + 4 * v, lbuf + 4 * v);

  // 2) per-d constants once per block (the only fp32 divides in the kernel)
  if (tid < DIMH) {
    float r = (f0[bt] * (float)(tid + 1)) / SRF;  // same op order as reference
    r = r - floorf(r);
    if (t == 0) r += rand_ini[b * DIMH + tid];
    sr[tid] = r;
    sb[tid] = base[(long)bt * DIMH + tid];
  }
  const float fv   = f0[bt];
  const float uvv  = (fv > 0.0f) ? 1.0f : 0.0f;
  const float namp = uvv * NOISE_STD + (1.0f - uvv) * (SINE_AMP / 3.0f);
  const float amp  = SINE_AMP * uvv;
  if (tid < upp)
    __builtin_nontemporal_store(uvv, uv_out + (long)bt * upp + tid);
  __syncthreads();                                // publish sr/sb

  // sine for 4 consecutive elements starting at j0 = 4*v
  auto sine4 = [&](int v) -> v4f {
    v4f s;
#pragma unroll
    for (int i = 0; i < 4; ++i) {
      const int j = 4 * v + i;
      const int u = j / DIMH;                     // const divisor -> mul/shift
      const int d = j - u * DIMH;
      const float r  = sr[d];
      const float bs = sb[d];
      const float kk = (float)(u + 1);
      const float hi = kk * r;                    // exact hi/lo split keeps
      const float lo = __builtin_fmaf(kk, r, -hi);//   mod-1 phase at fp32 ulp
      float ph = bs + (hi - floorf(hi)) + lo;
      ph = ph - floorf(ph);
      // v_sin_f32 takes revolutions: sin(2*pi*ph) in one instruction
      s[i] = __builtin_amdgcn_sinf(ph) * amp;
    }
    return s;
  };

  // 3) compute sines in registers while the async copy is in flight
  const int v0 = tid, v1 = tid + nth;
  v4f s0 = {}, s1 = {};
  if (v0 < nv4) s0 = sine4(v0);
  if (v1 < nv4) s1 = sine4(v1);

  // 4) each thread reads back exactly the LDS it loaded -> own-wave wait only
  wait_async0();

  const v4f* lb4 = (const v4f*)lbuf;
  v4f* so4 = (v4f*)(sine_out + foff);
  v4f* no4 = (v4f*)(noise_out + foff);
  if (v0 < nv4) {
    const v4f nz = lb4[v0] * namp;
    __builtin_nontemporal_store(nz, no4 + v0);
    __builtin_nontemporal_store(s0 + nz, so4 + v0);
  }
  if (v1 < nv4) {
    const v4f nz = lb4[v1] * namp;
    __builtin_nontemporal_store(nz, no4 + v1);
    __builtin_nontemporal_store(s1 + nz, so4 + v1);
  }
  // generic tail (absent for upp=256: nv4 == 2*nth exactly)
  for (int v = tid + 2 * nth; v < nv4; v += nth) {
    const v4f nz = lb4[v] * namp;
    __builtin_nontemporal_store(nz, no4 + v);
    __builtin_nontemporal_store(sine4(v) + nz, so4 + v);
  }
}

// ---------------------------------------------------------------------------
extern "C" void kernel_launch(void* const* d_in, const int* in_sizes, int n_in,
                              void* d_out, int out_size, void* d_ws, size_t ws_size,
                              hipStream_t stream) {
  const float* f0       = (const float*)d_in[0];
  const float* rand_ini = (const float*)d_in[1];
  const float* noise_z  = (const float*)d_in[2];

  const int  B   = in_sizes[1] / DIMH;          // 16
  const int  T   = in_sizes[0] / B;             // 800
  const long LD  = (long)in_sizes[2];           // B*L*DIM
  const long L   = LD / ((long)B * DIMH);       // 204800
  const int  upp = (int)(L / T);                // 256

  float* base      = (float*)d_ws;              // B*T*DIM floats (460 KB)
  float* sine_out  = (float*)d_out;
  float* uv_out    = sine_out + LD;
  float* noise_out = uv_out + (long)B * L;

  const int rows = B * DIMH;                    // 144
  sine_precompute<<<(rows + 127) / 128, 128, 0, stream>>>(
      f0, rand_ini, base, B, T, upp);

  // blockDim: two float4s per thread; >= upp (uv store) and >= DIMH
  const int chunk = upp * DIMH;                 // 2304 floats
  const int nv4   = chunk >> 2;                 // 576
  int nth = (nv4 + 1) / 2;                      // 288
  nth = (nth + 31) & ~31;                       // wave32 multiple
  if (nth < upp) nth = upp;
  const size_t shmem = (size_t)chunk * sizeof(float);
  sine_main<<<B * T, nth, shmem, stream>>>(
      f0, rand_ini, noise_z, base, sine_out, uv_out, noise_out, T, upp);
}